// SGGM_59949153517801
// MI455X (gfx1250) — compile-verified
//
#include <hip/hip_runtime.h>
#include <hip/hip_bf16.h>

#define BB 8
#define NN 128
#define HH 128
#define OUTC 768   // 6H
#define INC  384   // 3H
#define NT   16    // node types
#define NE   5     // edge types
#define OTILES 48  // 768 / 16

typedef __attribute__((ext_vector_type(2))) float v2f;
typedef __attribute__((ext_vector_type(8))) float v8f;

// ---------------------------------------------------------------------------
// Kernel 1: precompute
//   P[0][t][o] = W1 . embed_node[t]
//   P[1][t][o] = W2 . embed_node[t]
//   P[2][e][o] = W3 . embed_edge[e] + bias[o]     (rows e>=5 zero-padded)
// via V_WMMA_F32_16X16X4_F32. One wave (32 threads) per (table, o-tile) job.
// A layout (16x4 f32): lanes 0-15 -> M=lane, {K0,K1}; lanes 16-31 -> {K2,K3}.
// B layout (4x16 f32): lanes 0-15 -> N=lane, {K0,K1}; lanes 16-31 -> {K2,K3}.
// D layout (16x16 f32, 8 VGPR): lanes 0-15 -> M=vg, N=lane; lanes 16-31 -> M=vg+8.
//
// The zero-padded embedding tile is staged into LDS once so the hot K-loop
// has no divergent predication: each step is ds_load_b64 + global_load_b64
// + v_wmma, with two independent accumulators to overlap WMMA latency.
// ---------------------------------------------------------------------------
__global__ void SGGM_precompute_wmma(const float* __restrict__ embed_node,
                                     const float* __restrict__ embed_edge,
                                     const float* __restrict__ W,     // [768][384]
                                     const float* __restrict__ bias,  // [768]
                                     float* __restrict__ P) {         // [3][16][768]
    __shared__ __align__(16) float ldsE[NT * HH];   // 8 KB, zero-padded A tile

    const int job   = blockIdx.x;       // 0..143
    const int table = job / OTILES;     // 0,1,2
    const int o0    = (job % OTILES) * 16;

    const int lane  = threadIdx.x;      // wave32, EXEC all ones
    const int m     = lane & 15;        // row (A) / col (B) index
    const int khalf = (lane >> 4) * 2;  // 0 or 2

    const float* E     = (table == 2) ? embed_edge : embed_node;
    const int    erows = (table == 2) ? NE : NT;
    const int    coff  = table * HH;    // which 128-col slab of W_msg

    // Stage zero-padded E into LDS (divergence confined to this one loop).
    for (int idx = lane; idx < NT * HH; idx += 32) {
        const int r = idx >> 7;
        ldsE[idx] = (r < erows) ? E[idx] : 0.0f;
    }
    __syncthreads();

    const float* aptr = ldsE + m * HH + khalf;                 // ds_load_b64
    const float* bptr = W + (o0 + m) * INC + coff + khalf;     // global_load_b64

    v8f acc0 = {};
    v8f acc1 = {};
#pragma unroll 4
    for (int k = 0; k < HH; k += 8) {
        const v2f a0 = *(const v2f*)(aptr + k);
        const v2f b0 = *(const v2f*)(bptr + k);
        acc0 = __builtin_amdgcn_wmma_f32_16x16x4_f32(
                   false, a0, false, b0, (short)0, acc0, false, false);
        const v2f a1 = *(const v2f*)(aptr + k + 4);
        const v2f b1 = *(const v2f*)(bptr + k + 4);
        acc1 = __builtin_amdgcn_wmma_f32_16x16x4_f32(
                   false, a1, false, b1, (short)0, acc1, false, false);
    }
    acc0 += acc1;   // elementwise: D tiles share a layout

    const int nn    = lane & 15;
    const int mbase = (lane >> 4) * 8;
    float* outt = P + table * (NT * OUTC);
    const float badd = (table == 2) ? bias[o0 + nn] : 0.0f;
#pragma unroll
    for (int vg = 0; vg < 8; ++vg) {
        outt[(mbase + vg) * OUTC + (o0 + nn)] = acc0[vg] + badd;
    }
}

// ---------------------------------------------------------------------------
// Kernel 2: store-bound assembly.  One block per (b,i); 192 threads, each
// owns one float4 of the 768 output channels.  P2/Q tables in LDS, P1[x_i]
// slice in a register, x/adj rows broadcast from LDS, b128 coalesced stores.
// ---------------------------------------------------------------------------
#define TPB 192
__global__ void SGGM_assemble(const int* __restrict__ x,
                              const int* __restrict__ adj,
                              const float* __restrict__ P,   // [3][16][768]
                              float* __restrict__ out) {
    __shared__ __align__(16) float lds_P2[NT * OUTC];  // 48 KB
    __shared__ __align__(16) float lds_Q [NE * OUTC];  // 15 KB
    __shared__ int lds_xj[NN];
    __shared__ int lds_e [NN];

    const int bi = blockIdx.x;       // 0..B*N-1
    const int b  = bi >> 7;
    const int i  = bi & (NN - 1);
    const int tid = threadIdx.x;     // 0..191

    for (int idx = tid; idx < NT * OUTC; idx += TPB)
        lds_P2[idx] = P[NT * OUTC + idx];
    for (int idx = tid; idx < NE * OUTC; idx += TPB)
        lds_Q[idx] = P[2 * NT * OUTC + idx];
    if (tid < NN) {
        lds_xj[tid] = x[b * NN + tid];
        lds_e [tid] = adj[(b * NN + i) * NN + tid];
    }
    __syncthreads();

    const int xi = x[b * NN + i];
    const float4 p1 = ((const float4*)(P + xi * OUTC))[tid];

    float4* outp = (float4*)(out + (size_t)(b * NN + i) * NN * OUTC);
    for (int j = 0; j < NN; ++j) {
        const int e = lds_e[j];                        // uniform across block
        float4 r = ((const float4*)(lds_Q + e * OUTC))[tid];
        if (e > 0) {
            const int tj = lds_xj[j];
            const float4 p2 = ((const float4*)(lds_P2 + tj * OUTC))[tid];
            r.x += p1.x + p2.x;
            r.y += p1.y + p2.y;
            r.z += p1.z + p2.z;
            r.w += p1.w + p2.w;
        }
        outp[(size_t)j * (OUTC / 4) + tid] = r;
    }
}

// ---------------------------------------------------------------------------
extern "C" void kernel_launch(void* const* d_in, const int* in_sizes, int n_in,
                              void* d_out, int out_size, void* d_ws, size_t ws_size,
                              hipStream_t stream) {
    const int*   x          = (const int*)  d_in[0];   // [8,128]
    const int*   adj        = (const int*)  d_in[1];   // [8,128,128]
    const float* embed_node = (const float*)d_in[2];   // [16,128]
    const float* embed_edge = (const float*)d_in[3];   // [5,128]
    const float* W_msg      = (const float*)d_in[4];   // [768,384]
    const float* b_msg      = (const float*)d_in[5];   // [768]
    float*       out        = (float*)d_out;           // [8,128,128,768]
    float*       P          = (float*)d_ws;            // 3*16*768*4 = 147456 B

    // 3 tables * 48 o-tiles = 144 wave-jobs, one wave32 each.
    SGGM_precompute_wmma<<<3 * OTILES, 32, 0, stream>>>(embed_node, embed_edge,
                                                        W_msg, b_msg, P);
    // One block per (b,i) output row; store-bandwidth bound.
    SGGM_assemble<<<BB * NN, TPB, 0, stream>>>(x, adj, P, out);
}